// PredictSequences_12403865551019
// MI455X (gfx1250) — compile-verified
//
#include <hip/hip_runtime.h>
#include <math.h>

typedef __attribute__((ext_vector_type(2))) float v2f;
typedef __attribute__((ext_vector_type(8))) float v8f;

#define NROWS   500000
#define D0      128
#define UNITS   4
#define STACK   12
#define HFIN    (D0 + UNITS * STACK)   // 176
#define HS      180                    // padded LDS row stride in floats
#define NTILES  (NROWS / 16)           // 31250 16-row tiles (even)
#define WPB     4                      // waves per block (128 threads)
#define TPW     2                      // row-tiles per wave
#define WLDSF   7552                   // 7200 stacked-layer weights + 352 wf floats
#define SHOFF   WLDSF                  // BN shift table (48 floats) after weights
#define HOFF    (WLDSF + 48)           // h tiles after shift table (16B aligned)
#define BN_EPS  1e-3f

// Offset of layer i's weights in the compact LDS cache: sum_{j<i}(128+4j)*4
__device__ __forceinline__ int woff_layer(int i) { return 8 * i * i + 504 * i; }

// One K=4 chunk: load shared B once, issue one WMMA per row-tile.
#define CHUNK4(KOFF, WC, NCL, MSK, ACC0, ACC1)                                  \
    {                                                                           \
        v2f a0 = *reinterpret_cast<const v2f*>(h0 + aidx + (KOFF));             \
        v2f a1 = *reinterpret_cast<const v2f*>(h1 + aidx + (KOFF));             \
        v2f bw;                                                                 \
        bw.x = wlds[wo + ((KOFF) + khalf)     * (WC) + (NCL)] * (MSK);          \
        bw.y = wlds[wo + ((KOFF) + khalf + 1) * (WC) + (NCL)] * (MSK);          \
        ACC0 = __builtin_amdgcn_wmma_f32_16x16x4_f32(false, a0, false, bw,      \
                                                     (short)0, ACC0, false, false); \
        ACC1 = __builtin_amdgcn_wmma_f32_16x16x4_f32(false, a1, false, bw,      \
                                                     (short)0, ACC1, false, false); \
    }

__global__ void __launch_bounds__(128)
predict_stack_wmma(const float* __restrict__ x,
                   const float* __restrict__ w0,  const float* __restrict__ w1,
                   const float* __restrict__ w2,  const float* __restrict__ w3,
                   const float* __restrict__ w4,  const float* __restrict__ w5,
                   const float* __restrict__ w6,  const float* __restrict__ w7,
                   const float* __restrict__ w8,  const float* __restrict__ w9,
                   const float* __restrict__ w10, const float* __restrict__ w11,
                   const float* __restrict__ gamma, const float* __restrict__ beta,
                   const float* __restrict__ mean,  const float* __restrict__ var,
                   const float* __restrict__ wf,    const float* __restrict__ bf,
                   float* __restrict__ out)
{
    extern __shared__ float smem[];
    float* wlds  = smem;           // [WLDSF] BN-prescaled weights (+ raw wf)
    float* shlds = smem + SHOFF;   // [48] per-(layer,unit) BN shift
    float* hbase = smem + HOFF;    // [WPB][TPW][16][HS]

    const int tid = threadIdx.x;

    // ---- Stage weights with BN scale folded in; build shift table ----
    {
        const float* wsrc[STACK] = {w0,w1,w2,w3,w4,w5,w6,w7,w8,w9,w10,w11};
        #pragma unroll
        for (int i = 0; i < STACK; ++i) {
            const int off = woff_layer(i);
            const int cnt = (D0 + UNITS * i) * UNITS;
            for (int idx = tid; idx < cnt; idx += 128) {
                const int n  = idx & 3;
                const float s = gamma[i * UNITS + n] * rsqrtf(var[i * UNITS + n] + BN_EPS);
                wlds[off + idx] = wsrc[i][idx] * s;
            }
        }
        for (int idx = tid; idx < HFIN * 2; idx += 128)
            wlds[7200 + idx] = wf[idx];
        if (tid < STACK * UNITS) {
            const int n = tid & 3, i = tid >> 2;
            const float s = gamma[i * UNITS + n] * rsqrtf(var[i * UNITS + n] + BN_EPS);
            shlds[tid] = beta[tid] - mean[tid] * s;
        }
    }
    __syncthreads();

    const int wave = tid >> 5;
    const int lane = tid & 31;
    const int pair = blockIdx.x * WPB + wave;   // pair of 16-row tiles
    if (pair * TPW >= NTILES) return;           // uniform per wave

    float* h0 = hbase + (wave * TPW + 0) * 16 * HS;
    float* h1 = hbase + (wave * TPW + 1) * 16 * HS;

    // Per-lane WMMA coordinates (fp32 16x16x4, wave32):
    //   A: lane L<16 -> M=L, K={k,k+1}; L>=16 -> M=L-16, K={k+2,k+3}
    //   B: VGPR0 -> K = k + 2*(L>>4), N = L&15 (VGPR1: K+1)
    //   C/D: VGPR v -> M = v + 8*(L>>4), N = L&15
    const int mrow  = lane & 15;
    const int khalf = (lane >> 4) * 2;
    const int ncol  = lane & 15;
    const int mhi   = (lane >> 4) * 8;
    const bool n4   = (ncol < UNITS);
    const bool n2   = (ncol < 2);
    const float msk4 = n4 ? 1.0f : 0.0f;
    const float msk2 = n2 ? 1.0f : 0.0f;
    const int  ncl4  = n4 ? ncol : 0;
    const int  ncl2  = n2 ? ncol : 0;

    // ---- Load 32 consecutive rows of x into the two h tiles (cols 48..175) ----
    const int row0 = pair * TPW * 16;
    const float4* xv = reinterpret_cast<const float4*>(x);
    #pragma unroll 4
    for (int r = 0; r < 16; ++r) {
        float4 v0 = xv[(size_t)(row0 + r)      * (D0 / 4) + lane];
        float4 v1 = xv[(size_t)(row0 + 16 + r) * (D0 / 4) + lane];
        *reinterpret_cast<float4*>(&h0[r * HS + 48 + lane * 4]) = v0;
        *reinterpret_cast<float4*>(&h1[r * HS + 48 + lane * 4]) = v1;
    }

    // ---- 12 stacked dense(4) + BN + ReLU layers ----
    // Fully unrolled so K, base, wo and the odd-chunk peel are compile-time
    // constants per layer (no runtime accumulator-select merge code).
    #pragma unroll
    for (int i = 0; i < STACK; ++i) {
        const int base = 48 - 4 * i;
        const int K    = D0 + UNITS * i;        // multiple of 4; constant per layer
        const int wo   = woff_layer(i);
        const int aidx = mrow * HS + base + khalf;

        v8f cA0 = {0,0,0,0,0,0,0,0}, cA1 = {0,0,0,0,0,0,0,0};
        v8f cB0 = {0,0,0,0,0,0,0,0}, cB1 = {0,0,0,0,0,0,0,0};

        const int k0 = K & 4;                   // statically known after unroll
        if (k0) CHUNK4(0, UNITS, ncl4, msk4, cA0, cA1)
        #pragma unroll 2
        for (int k = k0; k < K; k += 8) {
            CHUNK4(k,     UNITS, ncl4, msk4, cA0, cA1)
            CHUNK4(k + 4, UNITS, ncl4, msk4, cB0, cB1)
        }
        const v8f c0 = cA0 + cB0;
        const v8f c1 = cA1 + cB1;

        // BN shift + ReLU (scale already folded into weights)
        if (n4) {
            const float sh = shlds[i * UNITS + ncol];
            const int ycol = base - UNITS;
            #pragma unroll
            for (int v = 0; v < 8; ++v) {
                const int M = v + mhi;
                h0[M * HS + ycol + ncol] = fmaxf(0.0f, c0[v] + sh);
                h1[M * HS + ycol + ncol] = fmaxf(0.0f, c1[v] + sh);
            }
        }
    }

    // ---- Final dense(2) over full h (K=176, 176%8==0) + softmax ----
    {
        const int wo   = 7200;
        const int aidx = mrow * HS + khalf;
        v8f cA0 = {0,0,0,0,0,0,0,0}, cA1 = {0,0,0,0,0,0,0,0};
        v8f cB0 = {0,0,0,0,0,0,0,0}, cB1 = {0,0,0,0,0,0,0,0};
        #pragma unroll 2
        for (int k = 0; k < HFIN; k += 8) {
            CHUNK4(k,     2, ncl2, msk2, cA0, cA1)
            CHUNK4(k + 4, 2, ncl2, msk2, cB0, cB1)
        }
        v8f cf[TPW];
        cf[0] = cA0 + cB0;
        cf[1] = cA1 + cB1;

        const float bfv = n2 ? bf[ncol] : 0.0f;
        #pragma unroll
        for (int t = 0; t < TPW; ++t) {
            const int trow = row0 + t * 16;
            #pragma unroll
            for (int v = 0; v < 8; ++v) {
                const float l = cf[t][v] + bfv;             // logit for column ncol
                const float o = __shfl_xor(l, 1, 32);       // partner logit (N 0<->1)
                const float m = fmaxf(l, o);
                const float e  = __expf(l - m);
                const float eo = __expf(o - m);
                const float p  = e / (e + eo);
                if (n2) {
                    const int M = v + mhi;
                    out[(size_t)(trow + M) * 2 + ncol] = p;
                }
            }
        }
    }
}

extern "C" void kernel_launch(void* const* d_in, const int* in_sizes, int n_in,
                              void* d_out, int out_size, void* d_ws, size_t ws_size,
                              hipStream_t stream)
{
    (void)in_sizes; (void)n_in; (void)out_size; (void)d_ws; (void)ws_size;
    const float* x   = (const float*)d_in[0];
    const float* w[12];
    for (int i = 0; i < 12; ++i) w[i] = (const float*)d_in[1 + i];
    const float* gamma = (const float*)d_in[13];
    const float* beta  = (const float*)d_in[14];
    const float* mean  = (const float*)d_in[15];
    const float* var   = (const float*)d_in[16];
    const float* wf    = (const float*)d_in[17];
    const float* bf    = (const float*)d_in[18];
    float* out = (float*)d_out;

    const int pairs  = NTILES / TPW;                          // 15625
    const int blocks = (pairs + WPB - 1) / WPB;               // 3907
    const size_t shmem =
        (size_t)(HOFF + WPB * TPW * 16 * HS) * sizeof(float); // ~120 KB

    predict_stack_wmma<<<dim3(blocks), dim3(128), shmem, stream>>>(
        x, w[0], w[1], w[2], w[3], w[4], w[5], w[6], w[7], w[8], w[9], w[10], w[11],
        gamma, beta, mean, var, wf, bf, out);
}